// FASTRGCN_19722489823543
// MI455X (gfx1250) — compile-verified
//
#include <hip/hip_runtime.h>
#include <cstdint>
#include <cstddef>

typedef __attribute__((ext_vector_type(2))) float v2f;
typedef __attribute__((ext_vector_type(8))) float v8f;

#define R_REL 16
#define DIN   128
#define NBASE 8

// ---------------------------------------------------------------------------
// W[r,i,o] = sum_b comp[r,b] * basis[b,i,o]     (tiny: <= 16*128*128 floats)
// ---------------------------------------------------------------------------
__global__ void make_w_kernel(const float* __restrict__ basis,
                              const float* __restrict__ comp,
                              float* __restrict__ W, int dout)
{
    int idx = blockIdx.x * blockDim.x + threadIdx.x;
    int total = R_REL * DIN * dout;
    if (idx >= total) return;
    int o = idx % dout;
    int i = (idx / dout) % DIN;
    int r = idx / (dout * DIN);
    float acc = 0.f;
#pragma unroll
    for (int b = 0; b < NBASE; ++b)
        acc += comp[r * NBASE + b] * basis[((size_t)b * DIN + i) * dout + o];
    W[idx] = acc;
}

// ---------------------------------------------------------------------------
// Edge scatter: S[(dst*R+rel), :] += h[src, :]; cnt[(dst*R+rel)] += 1
// One wave (32 lanes) per edge; each lane moves a float4 (128 floats total).
// ---------------------------------------------------------------------------
__global__ __launch_bounds__(256) void scatter_kernel(
    const float* __restrict__ h, const int* __restrict__ src,
    const int* __restrict__ dst, const int* __restrict__ et,
    float* __restrict__ S, float* __restrict__ cnt, int E)
{
    int e = (blockIdx.x * blockDim.x + threadIdx.x) >> 5;
    if (e >= E) return;
    int lane = threadIdx.x & 31;
    int s = src[e];
    int d = dst[e];
    int r = et[e];
    float4 v = ((const float4*)(h + (size_t)s * DIN))[lane];
    float* Sp = S + ((size_t)d * R_REL + r) * DIN + lane * 4;
    atomicAdd(Sp + 0, v.x);
    atomicAdd(Sp + 1, v.y);
    atomicAdd(Sp + 2, v.z);
    atomicAdd(Sp + 3, v.w);
    if (lane == 0) atomicAdd(cnt + (size_t)d * R_REL + r, 1.0f);
}

// ---------------------------------------------------------------------------
// out[n, o] = relu( sum_{r,k} S[n,r,k]/max(cnt[n,r],1) * W[r,k,o]
//                 + sum_k h[n,k]*root[k,o] + bias[o] )
// Wave = 16 nodes x full DO.  fp32 WMMA 16x16x4.
// B staged in LDS pre-swizzled into WMMA fragment order so each B fragment is
// one aligned ds_load_b64 (no register repacking):
//   Bs[(k>>2)*GSTR + ((k>>1)&1)*PSTR + col*2 + (k&1)]
// PSTR chosen so lane-halves hit disjoint LDS bank halves.
// ---------------------------------------------------------------------------
template<int DO, bool RELU>
__global__ __launch_bounds__(256) void rgcn_gemm_kernel(
    const float* __restrict__ S, const float* __restrict__ cnt,
    const float* __restrict__ W, const float* __restrict__ h_in,
    const float* __restrict__ root, const float* __restrict__ bias,
    float* __restrict__ h_out, int N, int nTiles)
{
    constexpr int NT   = DO / 16;                    // output column tiles
    constexpr int KC   = 64;                         // K rows staged per slab
    constexpr int PSTR = ((2 * DO) % 64 == 0) ? (2 * DO + 32) : (2 * DO); // ==32 mod 64
    constexpr int GSTR = 2 * PSTR;                   // stride per K-group of 4
    __shared__ float Bs[(KC / 4) * GSTR];

    const int lane  = threadIdx.x & 31;
    const int wv    = threadIdx.x >> 5;
    const int tile  = blockIdx.x * (blockDim.x >> 5) + wv;
    const int m     = lane & 15;                     // A row / B col within tile
    const int khalf = (lane >> 4) << 1;              // 0 or 2 (K half select)
    const int bbase = (lane >> 4) * PSTR + m * 2;    // lane's base into a K-group

    int row = (tile < nTiles) ? tile * 16 + m : 0;   // clamped load row
    if (row >= N) row = N - 1;

    v8f acc[NT] = {};

    // ---- relation part: S/cnt  x  W ----
    for (int rel = 0; rel < R_REL; ++rel) {
        const float cn  = cnt[(size_t)row * R_REL + rel];
        const float inv = 1.0f / fmaxf(cn, 1.0f);
        const float* __restrict__ sr = S + ((size_t)row * R_REL + rel) * DIN;
        const float* __restrict__ wr = W + (size_t)rel * DIN * DO;
        for (int kc = 0; kc < DIN; kc += KC) {
            __syncthreads();
            // stage KC x DO slab of W into fragment-swizzled LDS
            for (int idx = threadIdx.x; idx < (KC / 2) * DO; idx += blockDim.x) {
                int o  = idx % DO;
                int kp = idx / DO;                   // pair index, k = 2*kp
                int k  = 2 * kp;
                v2f p;
                p.x = wr[(size_t)(kc + k)     * DO + o];
                p.y = wr[(size_t)(kc + k + 1) * DO + o];
                *(v2f*)&Bs[(kp >> 1) * GSTR + (kp & 1) * PSTR + o * 2] = p;
            }
            __syncthreads();
#pragma unroll
            for (int kb = 0; kb < KC; kb += 4) {
                v2f a = *(const v2f*)&sr[kc + kb + khalf];
                a.x *= inv;
                a.y *= inv;
                const float* bg = &Bs[(kb >> 2) * GSTR + bbase];
#pragma unroll
                for (int t = 0; t < NT; ++t) {
                    v2f b = *(const v2f*)&bg[t * 32];
                    acc[t] = __builtin_amdgcn_wmma_f32_16x16x4_f32(
                        false, a, false, b, (short)0, acc[t], false, false);
                }
            }
        }
    }

    // ---- root part: h_in x root (extra K = DIN) ----
    const float* __restrict__ hr = h_in + (size_t)row * DIN;
    for (int kc = 0; kc < DIN; kc += KC) {
        __syncthreads();
        for (int idx = threadIdx.x; idx < (KC / 2) * DO; idx += blockDim.x) {
            int o  = idx % DO;
            int kp = idx / DO;
            int k  = 2 * kp;
            v2f p;
            p.x = root[(size_t)(kc + k)     * DO + o];
            p.y = root[(size_t)(kc + k + 1) * DO + o];
            *(v2f*)&Bs[(kp >> 1) * GSTR + (kp & 1) * PSTR + o * 2] = p;
        }
        __syncthreads();
#pragma unroll
        for (int kb = 0; kb < KC; kb += 4) {
            v2f a = *(const v2f*)&hr[kc + kb + khalf];
            const float* bg = &Bs[(kb >> 2) * GSTR + bbase];
#pragma unroll
            for (int t = 0; t < NT; ++t) {
                v2f b = *(const v2f*)&bg[t * 32];
                acc[t] = __builtin_amdgcn_wmma_f32_16x16x4_f32(
                    false, a, false, b, (short)0, acc[t], false, false);
            }
        }
    }

    // ---- epilogue: bias, relu, store ----
    if (tile < nTiles) {
        const int n0 = tile * 16;
#pragma unroll
        for (int t = 0; t < NT; ++t) {
            const int col = t * 16 + m;
            const float bi = bias[col];
#pragma unroll
            for (int v = 0; v < 8; ++v) {
                const int rr = n0 + v + ((lane >> 4) << 3);
                if (rr < N) {
                    float val = acc[t][v] + bi;
                    if (RELU) val = fmaxf(val, 0.f);
                    h_out[(size_t)rr * DO + col] = val;
                }
            }
        }
    }
}

// ---------------------------------------------------------------------------
// Host side
// ---------------------------------------------------------------------------
extern "C" void kernel_launch(void* const* d_in, const int* in_sizes, int n_in,
                              void* d_out, int out_size, void* d_ws, size_t ws_size,
                              hipStream_t stream)
{
    const float* x  = (const float*)d_in[0];
    const int*   ei = (const int*)d_in[1];
    const int*   et = (const int*)d_in[2];
    const int N = in_sizes[0] / DIN;
    const int E = in_sizes[1] / 2;
    const int* src = ei;
    const int* dst = ei + E;

    // per-layer params: d_in[3+4l .. 6+4l] = basis, comp, root, bias
    const float* basis[3] = { (const float*)d_in[3], (const float*)d_in[7],  (const float*)d_in[11] };
    const float* comp [3] = { (const float*)d_in[4], (const float*)d_in[8],  (const float*)d_in[12] };
    const float* rootp[3] = { (const float*)d_in[5], (const float*)d_in[9],  (const float*)d_in[13] };
    const float* biasp[3] = { (const float*)d_in[6], (const float*)d_in[10], (const float*)d_in[14] };

    // workspace carve-up
    float* S    = (float*)d_ws;                         // N*R*DIN
    float* cntf = S    + (size_t)N * R_REL * DIN;       // N*R
    float* W    = cntf + (size_t)N * R_REL;             // R*DIN*DIN (max)
    float* h1   = W    + (size_t)R_REL * DIN * DIN;     // N*DIN
    float* h2   = h1   + (size_t)N * DIN;               // N*DIN

    const int nTiles = (N + 15) / 16;
    dim3 gemmGrid((nTiles + 7) / 8);
    dim3 scatGrid((unsigned)(((size_t)E * 32 + 255) / 256));

    const size_t Sbytes = (size_t)N * R_REL * DIN * sizeof(float);
    const size_t Cbytes = (size_t)N * R_REL * sizeof(float);

    // ----- layer 0: x -> h1 (128 -> 128, relu) -----
    hipMemsetAsync(S,    0, Sbytes, stream);
    hipMemsetAsync(cntf, 0, Cbytes, stream);
    make_w_kernel<<<(R_REL * DIN * DIN + 255) / 256, 256, 0, stream>>>(basis[0], comp[0], W, DIN);
    scatter_kernel<<<scatGrid, 256, 0, stream>>>(x, src, dst, et, S, cntf, E);
    rgcn_gemm_kernel<128, true><<<gemmGrid, 256, 0, stream>>>(
        S, cntf, W, x, rootp[0], biasp[0], h1, N, nTiles);

    // ----- layer 1: h1 -> h2 (128 -> 128, relu) -----
    hipMemsetAsync(S,    0, Sbytes, stream);
    hipMemsetAsync(cntf, 0, Cbytes, stream);
    make_w_kernel<<<(R_REL * DIN * DIN + 255) / 256, 256, 0, stream>>>(basis[1], comp[1], W, DIN);
    scatter_kernel<<<scatGrid, 256, 0, stream>>>(h1, src, dst, et, S, cntf, E);
    rgcn_gemm_kernel<128, true><<<gemmGrid, 256, 0, stream>>>(
        S, cntf, W, h1, rootp[1], biasp[1], h2, N, nTiles);

    // ----- layer 2: h2 -> out (128 -> 16, no relu) -----
    hipMemsetAsync(S,    0, Sbytes, stream);
    hipMemsetAsync(cntf, 0, Cbytes, stream);
    make_w_kernel<<<(R_REL * DIN * 16 + 255) / 256, 256, 0, stream>>>(basis[2], comp[2], W, 16);
    scatter_kernel<<<scatGrid, 256, 0, stream>>>(h2, src, dst, et, S, cntf, E);
    rgcn_gemm_kernel<16, false><<<gemmGrid, 256, 0, stream>>>(
        S, cntf, W, h2, rootp[2], biasp[2], (float*)d_out, N, nTiles);
}